// GAT_9620726743401
// MI455X (gfx1250) — compile-verified
//
#include <hip/hip_runtime.h>

// ---------------------------------------------------------------------------
// 2-layer GAT (PyG GATConv semantics) for MI455X / gfx1250, wave32 + WMMA.
// GEMMs use V_WMMA_F32_16X16X4_F32: fp32 in/out matches the reference, and at
// 23.3 TB/s the GEMMs (51 MB / 26 MB of traffic) are memory-bound, so the f32
// matrix pipe is nowhere near the limiter. W is staged in LDS pre-packed as
// float2-per-(k-pair, n) so every B fragment is one aligned ds_load_b64.
// ---------------------------------------------------------------------------

typedef __attribute__((ext_vector_type(2))) float v2f;
typedef __attribute__((ext_vector_type(8))) float v8f;

#define NEG_SLOPE 0.2f
#define NEG_BIG  -3.402823466e38f

__device__ __forceinline__ float lrelu(float x) {
    return x > 0.f ? x : NEG_SLOPE * x;
}

__device__ __forceinline__ void atomicMaxF(float* addr, float val) {
    unsigned int* ua = (unsigned int*)addr;
    unsigned int old = *ua;
    while (__uint_as_float(old) < val) {
        unsigned int assumed = old;
        old = atomicCAS(ua, assumed, __float_as_uint(val));
        if (old == assumed) break;
    }
}

// ---------------------------------------------------------------------------
// GEMM: Y[n_rows x NOUT] = X[n_rows x 128] * W[128 x NOUT]   (NOUT = 128 or 64)
// Workgroup = 256 threads = 8 wave32; each wave computes a 16-row strip across
// all NOUT columns as NOUT/16 WMMA 16x16 f32 accumulator tiles.
// LDS image of W is packed: Wl[(k>>1)*NOUT + n] = float2(W[k][n], W[k+1][n]),
// k even — so a B fragment (lane needs W[kk][n], W[kk+1][n], kk even) is one
// 64-bit DS load into an even-aligned VGPR pair, no repacking moves.
// A/B fragment lane layout per CDNA5 ISA 7.12.2 (32-bit 16x4 / 4x16):
//   lanes 0-15 hold K = k0,k0+1 ; lanes 16-31 hold K = k0+2,k0+3.
// ---------------------------------------------------------------------------
template <int NOUT>
__global__ __launch_bounds__(256)
void gat_gemm_wmma(const float* __restrict__ X, const float* __restrict__ W,
                   float* __restrict__ Y, int n_rows)
{
    __shared__ float2 Wl[64 * NOUT];         // 128*NOUT floats, packed by k-pair
    const int tid = threadIdx.x;
    for (int i = tid; i < 128 * NOUT; i += 256) {
        const int k = i / NOUT, n = i - k * NOUT;
        // scatter into packed layout: pair index (k>>1), element (k&1)
        ((float*)&Wl[(k >> 1) * NOUT + n])[k & 1] = W[i];
    }
    __syncthreads();

    const int wave = tid >> 5;
    const int lane = tid & 31;
    const int l16  = lane & 15;
    const int hi   = lane >> 4;              // 0: K,K+1   1: K+2,K+3
    const int rowBase = blockIdx.x * 128 + wave * 16;
    if (rowBase + 16 > n_rows) return;       // n_rows % 16 == 0 here

    constexpr int NT = NOUT / 16;
    v8f zero = {};
    v8f acc[NT];
#pragma unroll
    for (int t = 0; t < NT; ++t) acc[t] = zero;

    const float* xrow = X + (size_t)rowBase * 128;

#pragma unroll 4
    for (int k0 = 0; k0 < 128; k0 += 4) {
        const int kk = k0 + hi * 2;          // even
        const float2 av = *(const float2*)(xrow + l16 * 128 + kk);  // 8B aligned
        v2f a; a.x = av.x; a.y = av.y;       // A[M=l16][kk], A[M=l16][kk+1]
        const float2* wrow = &Wl[(kk >> 1) * NOUT];
#pragma unroll
        for (int t = 0; t < NT; ++t) {
            const float2 bv = wrow[t * 16 + l16];   // (W[kk][n], W[kk+1][n])
            v2f b; b.x = bv.x; b.y = bv.y;
            acc[t] = __builtin_amdgcn_wmma_f32_16x16x4_f32(
                false, a, false, b, (short)0, acc[t], false, false);
        }
    }

    // C/D layout: VGPR v, lanes 0-15 -> M=v, lanes 16-31 -> M=v+8; N=l16.
#pragma unroll
    for (int t = 0; t < NT; ++t) {
#pragma unroll
        for (int v = 0; v < 8; ++v) {
            Y[(size_t)(rowBase + v + hi * 8) * NOUT + t * 16 + l16] = acc[t][v];
        }
    }
}

// ---------------------------------------------------------------------------
// Per-node attention scores: a_s[n,h] = dot(xp[n,h,:], att_src[h,:]) (C per head)
// One wave per node, shfl_xor tree reduction (wave32).
// ---------------------------------------------------------------------------
template <int C>
__global__ __launch_bounds__(128)
void gat_att_scores(const float* __restrict__ XP, const float* __restrict__ att_s,
                    const float* __restrict__ att_d, float* __restrict__ a_s,
                    float* __restrict__ a_d, int n)
{
    const int wave = threadIdx.x >> 5, lane = threadIdx.x & 31;
    const int node = blockIdx.x * 4 + wave;
    if (node >= n) return;
    const float* row = XP + (size_t)node * (2 * C);
#pragma unroll
    for (int h = 0; h < 2; ++h) {
        float ps = 0.f, pd = 0.f;
#pragma unroll
        for (int c = lane; c < C; c += 32) {
            const float v = row[h * C + c];
            ps += v * att_s[h * C + c];
            pd += v * att_d[h * C + c];
        }
#pragma unroll
        for (int off = 16; off > 0; off >>= 1) {
            ps += __shfl_xor(ps, off, 32);
            pd += __shfl_xor(pd, off, 32);
        }
        if (lane == 0) { a_s[node * 2 + h] = ps; a_d[node * 2 + h] = pd; }
    }
}

// ---------------------------------------------------------------------------
// Init: zero accumulator (accN floats), set emax = -BIG, denom = 0 (nh entries).
// ---------------------------------------------------------------------------
__global__ void gat_init(float* __restrict__ acc, long accN,
                         float* __restrict__ emax, float* __restrict__ denom, long nh)
{
    const long i = (long)blockIdx.x * blockDim.x + threadIdx.x;
    if (i < accN) acc[i] = 0.f;
    if (i < nh) { emax[i] = NEG_BIG; denom[i] = 0.f; }
}

// ---------------------------------------------------------------------------
// Edge pass 1: segment max of leaky_relu(a_s[src]+a_d[dst]) per (dst, head).
// Edge ids >= E are self loops (src = dst = id - E), matching the reference's
// concatenated [edge_index | self_loops].
// ---------------------------------------------------------------------------
__global__ void gat_edge_max(const float* __restrict__ a_s, const float* __restrict__ a_d,
                             const int* __restrict__ src, const int* __restrict__ dst,
                             int E, int n, float* __restrict__ emax)
{
    const int i = blockIdx.x * blockDim.x + threadIdx.x;
    const int Etot = E + n;
    if (i >= Etot * 2) return;
    const int e = i >> 1, h = i & 1;
    int s, d;
    if (e < E) { s = src[e]; d = dst[e]; } else { s = d = e - E; }
    const float v = lrelu(a_s[s * 2 + h] + a_d[d * 2 + h]);
    atomicMaxF(&emax[d * 2 + h], v);
}

// Edge pass 2: ee = exp(e - emax[dst]); stash per-edge, atomic-add denominator.
__global__ void gat_edge_expsum(const float* __restrict__ a_s, const float* __restrict__ a_d,
                                const float* __restrict__ emax,
                                const int* __restrict__ src, const int* __restrict__ dst,
                                int E, int n, float* __restrict__ ee,
                                float* __restrict__ denom)
{
    const int i = blockIdx.x * blockDim.x + threadIdx.x;
    const int Etot = E + n;
    if (i >= Etot * 2) return;
    const int e = i >> 1, h = i & 1;
    int s, d;
    if (e < E) { s = src[e]; d = dst[e]; } else { s = d = e - E; }
    const float v  = lrelu(a_s[s * 2 + h] + a_d[d * 2 + h]);
    const float ex = __expf(v - emax[d * 2 + h]);
    ee[i] = ex;
    atomicAdd(&denom[d * 2 + h], ex);
}

// ---------------------------------------------------------------------------
// Edge pass 3: acc[dst] += xp[src] * alpha  (alpha = ee / denom[dst], per head).
// One wave per edge; all 32 lanes active, each moving WIDTH/32 floats of the
// message row (float4 for WIDTH=128, float2 for WIDTH=64) and issuing
// global_atomic_add_f32 per element. Source row prefetched (global_prefetch_b8)
// before the alpha division to hide gather latency — this phase dominates the
// runtime (~1.3 GB of L2/HBM traffic across both layers).
// ---------------------------------------------------------------------------
template <int WIDTH>
__global__ __launch_bounds__(128)
void gat_edge_scatter(const float* __restrict__ XP, const float* __restrict__ ee,
                      const float* __restrict__ denom,
                      const int* __restrict__ src, const int* __restrict__ dst,
                      int E, int n, float* __restrict__ acc)
{
    const int wave = threadIdx.x >> 5, lane = threadIdx.x & 31;
    const int e = blockIdx.x * 4 + wave;
    const int Etot = E + n;
    if (e >= Etot) return;
    int s, d;
    if (e < E) { s = src[e]; d = dst[e]; } else { s = d = e - E; }

    constexpr int VEC = WIDTH / 32;                  // floats per lane: 4 or 2
    const float* xr = XP + (size_t)s * WIDTH + lane * VEC;
    __builtin_prefetch(xr, 0, 3);                    // global_prefetch_b8

    const int   h     = (lane * VEC) / (WIDTH / 2);  // head of this lane's span
    const float alpha = ee[e * 2 + h] / denom[d * 2 + h];
    float* ap = acc + (size_t)d * WIDTH + lane * VEC;

    if constexpr (VEC == 4) {
        const float4 xv = *(const float4*)xr;
        atomicAdd(ap + 0, xv.x * alpha);
        atomicAdd(ap + 1, xv.y * alpha);
        atomicAdd(ap + 2, xv.z * alpha);
        atomicAdd(ap + 3, xv.w * alpha);
    } else {
        const float2 xv = *(const float2*)xr;
        atomicAdd(ap + 0, xv.x * alpha);
        atomicAdd(ap + 1, xv.y * alpha);
    }
}

// Epilogue 1: h = elu(acc + bias), in place.
__global__ void gat_bias_elu(float* __restrict__ h, const float* __restrict__ b,
                             long total, int width)
{
    const long i = (long)blockIdx.x * blockDim.x + threadIdx.x;
    if (i >= total) return;
    const float v = h[i] + b[i % width];
    h[i] = v > 0.f ? v : expm1f(v);
}

// Epilogue 2: out = log_softmax(acc + bias) over 64 channels; wave per node.
__global__ __launch_bounds__(128)
void gat_logsoftmax(const float* __restrict__ acc, const float* __restrict__ b,
                    float* __restrict__ out, int n)
{
    const int wave = threadIdx.x >> 5, lane = threadIdx.x & 31;
    const int node = blockIdx.x * 4 + wave;
    if (node >= n) return;
    const float* row = acc + (size_t)node * 64;
    const float v0 = row[lane]      + b[lane];
    const float v1 = row[lane + 32] + b[lane + 32];
    float m = fmaxf(v0, v1);
#pragma unroll
    for (int off = 16; off > 0; off >>= 1) m = fmaxf(m, __shfl_xor(m, off, 32));
    float sum = __expf(v0 - m) + __expf(v1 - m);
#pragma unroll
    for (int off = 16; off > 0; off >>= 1) sum += __shfl_xor(sum, off, 32);
    const float ls = __logf(sum);
    out[(size_t)node * 64 + lane]      = v0 - m - ls;
    out[(size_t)node * 64 + lane + 32] = v1 - m - ls;
}

// ---------------------------------------------------------------------------
extern "C" void kernel_launch(void* const* d_in, const int* in_sizes, int n_in,
                              void* d_out, int out_size, void* d_ws, size_t ws_size,
                              hipStream_t stream)
{
    const float* x    = (const float*)d_in[0];   // [N,128]
    const int*   ei   = (const int*)  d_in[1];   // [2,E]
    const float* W1   = (const float*)d_in[2];   // [128,128]
    const float* a1sW = (const float*)d_in[3];   // [2,64]
    const float* a1dW = (const float*)d_in[4];   // [2,64]
    const float* b1   = (const float*)d_in[5];   // [128]
    const float* W2   = (const float*)d_in[6];   // [128,64]
    const float* a2sW = (const float*)d_in[7];   // [2,32]
    const float* a2dW = (const float*)d_in[8];   // [2,32]
    const float* b2   = (const float*)d_in[9];   // [64]
    float* out = (float*)d_out;

    const int N    = in_sizes[0] / 128;
    const int E    = in_sizes[1] / 2;
    const int Etot = E + N;
    const int* src = ei;
    const int* dst = ei + E;

    // ---- workspace carving (floats) ----
    float* ws   = (float*)d_ws;
    float* xp1  = ws;  ws += (size_t)N * 128;   // conv1 transformed features
    float* h    = ws;  ws += (size_t)N * 128;   // conv1 aggregate -> elu -> conv2 input
    float* xp2  = ws;  ws += (size_t)N * 64;    // conv2 transformed features
    float* acc2 = ws;  ws += (size_t)N * 64;    // conv2 aggregate
    float* as1  = ws;  ws += (size_t)N * 2;
    float* ad1  = ws;  ws += (size_t)N * 2;
    float* em1  = ws;  ws += (size_t)N * 2;
    float* dn1  = ws;  ws += (size_t)N * 2;
    float* as2  = ws;  ws += (size_t)N * 2;
    float* ad2  = ws;  ws += (size_t)N * 2;
    float* em2  = ws;  ws += (size_t)N * 2;
    float* dn2  = ws;  ws += (size_t)N * 2;
    float* ee   = ws;  ws += (size_t)Etot * 2;  // per-(edge,head) exp, reused per layer

    const int gemmBlocks = (N + 127) / 128;     // 256 thr = 8 waves = 128 rows/block
    const int nodeWaves  = (N + 3) / 4;
    const int edgeThr    = (Etot * 2 + 255) / 256;
    const int edgeWaves  = (Etot + 3) / 4;

    // ---------------- layer 1 ----------------
    {
        const long accN = (long)N * 128, nh = (long)N * 2;
        gat_init<<<(int)((accN + 255) / 256), 256, 0, stream>>>(h, accN, em1, dn1, nh);
    }
    gat_gemm_wmma<128><<<gemmBlocks, 256, 0, stream>>>(x, W1, xp1, N);
    gat_att_scores<64><<<nodeWaves, 128, 0, stream>>>(xp1, a1sW, a1dW, as1, ad1, N);
    gat_edge_max    <<<edgeThr, 256, 0, stream>>>(as1, ad1, src, dst, E, N, em1);
    gat_edge_expsum <<<edgeThr, 256, 0, stream>>>(as1, ad1, em1, src, dst, E, N, ee, dn1);
    gat_edge_scatter<128><<<edgeWaves, 128, 0, stream>>>(xp1, ee, dn1, src, dst, E, N, h);
    {
        const long total = (long)N * 128;
        gat_bias_elu<<<(int)((total + 255) / 256), 256, 0, stream>>>(h, b1, total, 128);
    }

    // ---------------- layer 2 ----------------
    {
        const long accN = (long)N * 64, nh = (long)N * 2;
        gat_init<<<(int)((accN + 255) / 256), 256, 0, stream>>>(acc2, accN, em2, dn2, nh);
    }
    gat_gemm_wmma<64><<<gemmBlocks, 256, 0, stream>>>(h, W2, xp2, N);
    gat_att_scores<32><<<nodeWaves, 128, 0, stream>>>(xp2, a2sW, a2dW, as2, ad2, N);
    gat_edge_max    <<<edgeThr, 256, 0, stream>>>(as2, ad2, src, dst, E, N, em2);
    gat_edge_expsum <<<edgeThr, 256, 0, stream>>>(as2, ad2, em2, src, dst, E, N, ee, dn2);
    gat_edge_scatter<64><<<edgeWaves, 128, 0, stream>>>(xp2, ee, dn2, src, dst, E, N, acc2);

    gat_logsoftmax<<<nodeWaves, 128, 0, stream>>>(acc2, b2, out, N);
    (void)n_in; (void)out_size; (void)ws_size;
}